// SPDUnVectorize_21328807592136
// MI455X (gfx1250) — compile-verified
//
#include <hip/hip_runtime.h>
#include <stdint.h>

// Problem geometry (fixed by the reference): n = 256, L = n(n+1)/2 = 32896.
#define NDIM     256
#define LPACK    32896              // packed upper-triangle length per batch
#define L4       (LPACK / 4)        // 8224 float4 chunks
#define OUT_PER_B (NDIM * NDIM)     // 65536 floats per batch
#define THREADS  1024
#define LDS_BYTES (LPACK * 4)       // 131,584 B — legal on CDNA5 (320 KB/WGP)

typedef int v4i __attribute__((vector_size(16)));
typedef __attribute__((address_space(1))) v4i* gv4i_p;   // global int4*
typedef __attribute__((address_space(3))) v4i* lv4i_p;   // LDS int4*

__global__ __launch_bounds__(THREADS)
void spd_unvec_kernel(const float* __restrict__ in, float* __restrict__ out) {
    extern __shared__ float smem[];   // LPACK floats
    const int b   = blockIdx.x;
    const int tid = threadIdx.x;

    const float* gsrc = in + (size_t)b * LPACK;

    // ---- Stage: async global -> LDS, b128 per lane (coalesced 512B/wave) ----
#if __has_builtin(__builtin_amdgcn_global_load_async_to_lds_b128)
    for (int t = tid; t < L4; t += THREADS) {
        __builtin_amdgcn_global_load_async_to_lds_b128(
            (gv4i_p)(gsrc + 4 * t),
            (lv4i_p)(smem + 4 * t),
            /*offset=*/0, /*cpol=*/0);
    }
#if __has_builtin(__builtin_amdgcn_s_wait_asynccnt)
    __builtin_amdgcn_s_wait_asynccnt(0);
#else
    asm volatile("s_wait_asynccnt 0" ::: "memory");
#endif
#else
    for (int t = tid; t < L4; t += THREADS) {
        unsigned ldsaddr = (unsigned)(uintptr_t)(smem + 4 * t);
        unsigned goff    = (unsigned)t * 16u;
        asm volatile("global_load_async_to_lds_b128 %0, %1, %2"
                     :: "v"(ldsaddr), "v"(goff), "s"(gsrc)
                     : "memory");
    }
    asm volatile("s_wait_asynccnt 0" ::: "memory");
#endif
    __syncthreads();

    // ---- Emit: symmetric gather from LDS, fully coalesced float4 stores ----
    //
    // idx = tid + w*1024  =>  row i = (tid>>6) + 16*w,  col block j0 = (tid&63)*4
    // upper (i <= j):  k = [i*N - i(i-1)/2 - i + j0] + e   (rbj per iteration)
    // lower (i >  j):  k = [j*N - j(j-1)/2 - j] + i        (C[e] loop-invariant)
    const int j0 = (tid & 63) << 2;
    const int i0 = tid >> 6;

    int C[4];
#pragma unroll
    for (int e = 0; e < 4; ++e) {
        int j = j0 + e;
        C[e] = j * NDIM - ((j * (j - 1)) >> 1) - j;
    }

    float4* out4 = (float4*)(out + (size_t)b * OUT_PER_B);
#pragma unroll
    for (int w = 0; w < (OUT_PER_B / 4) / THREADS; ++w) {   // 16 iters
        const int i   = i0 + (w << 4);
        const int d   = i - j0;                              // e >= d -> upper
        const int rbj = i * NDIM - ((i * (i - 1)) >> 1) - i + j0;
        float4 v;
        float* vp = (float*)&v;
#pragma unroll
        for (int e = 0; e < 4; ++e) {
            int k = (d <= e) ? (rbj + e) : (C[e] + i);
            vp[e] = smem[k];
        }
        out4[tid + (w << 10)] = v;
    }
}

extern "C" void kernel_launch(void* const* d_in, const int* in_sizes, int n_in,
                              void* d_out, int out_size, void* d_ws, size_t ws_size,
                              hipStream_t stream) {
    (void)n_in; (void)d_ws; (void)ws_size; (void)out_size;
    const float* in  = (const float*)d_in[0];
    float*       out = (float*)d_out;

    const int B = in_sizes[0] / LPACK;   // 1024

    // Dynamic LDS > 64KB needs the opt-in attribute (idempotent, graph-safe).
    (void)hipFuncSetAttribute(reinterpret_cast<const void*>(spd_unvec_kernel),
                              hipFuncAttributeMaxDynamicSharedMemorySize, LDS_BYTES);

    spd_unvec_kernel<<<B, THREADS, LDS_BYTES, stream>>>(in, out);
}